// NeuralCDE_60636348284936
// MI455X (gfx1250) — compile-verified
//
#include <hip/hip_runtime.h>
#include <math.h>

// Neural CDE RK4 scan for MI455X (gfx1250), wave32, f32 WMMA 16x16x4.
// B=512, L=512, C=16, H=32. One workgroup per 16-row batch tile (grid=32),
// 8 waves; wave w owns output columns [64w, 64w+64) = h in [4w, 4w+4), all c.

typedef float v2f __attribute__((ext_vector_type(2)));
typedef float v8f __attribute__((ext_vector_type(8)));

#define B_   512
#define L_   512
#define C_   16
#define H_   32
#define HC_  (H_ * C_)   // 512
#define ZS   34          // z / k row stride (even -> 8B-aligned b64, conflict-free)
#define DXS  20          // dX row stride (16B-aligned b128 rows)
#define FS   20          // F-tile row stride (16B-aligned b128 rows)

// Branch-free tanh. gfx1250 has hardware V_TANH_F32 (single TRANS op that
// co-executes with VALU); fall back to exp2+rcp (also branch-free, saturates
// to +-1, propagates NaN) if the builtin is not declared by this toolchain.
__device__ __forceinline__ float fast_tanh(float x) {
#if __has_builtin(__builtin_amdgcn_tanhf)
    return __builtin_amdgcn_tanhf(x);
#else
    // tanh(x) = 1 - 2/(exp(2x)+1);  exp(2x) = exp2(x * 2*log2(e))
    float e = __builtin_amdgcn_exp2f(x * 2.885390081777927f);
    return 1.0f - 2.0f * __builtin_amdgcn_rcpf(e + 1.0f);
#endif
}

__global__ __launch_bounds__(256, 1)
void cde_rk4_kernel(const float* __restrict__ X,
                    const float* __restrict__ Wi,   // (C,H)
                    const float* __restrict__ bi,   // (H)
                    const float* __restrict__ Wf,   // (H, H*C)
                    const float* __restrict__ bf,   // (H*C)
                    float* __restrict__ out)        // (B,H)
{
    __shared__ float zb[16 * ZS];          // z   (16 x 32, padded)
    __shared__ float kb[4][16 * ZS];       // k1..k4
    __shared__ float dbuf[2][16 * DXS];    // diff double-buffer
    __shared__ float gmb[16 * DXS];        // gm
    __shared__ float xcur[16 * 17];        // X[:, t, :]
    __shared__ float fb[8][16 * FS];       // per-wave tanh(F) tile (16 x 64)

    const int tid  = threadIdx.x;
    const int lane = tid & 31;
    const int w    = tid >> 5;       // wave id 0..7
    const int lo   = lane & 15;
    const int hi   = lane >> 4;
    const int gb0  = blockIdx.x * 16;
    float* const fbw = &fb[w][0];

    // ---- preload this wave's W_f fragments (resident for all 2044 GEMMs) ----
    // B 4x16 f32 frag: lane lo = column N, VGPR v = row k0 + 2*hi + v.
    v2f   bfrag[4][8];
    float bfr[4];
    #pragma unroll
    for (int tt = 0; tt < 4; ++tt) {
        const int jb = 64 * w + 16 * tt + lo;       // global column in [0,512)
        bfr[tt] = bf[jb];
        #pragma unroll
        for (int kk = 0; kk < 8; ++kk) {
            const int r = 4 * kk + 2 * hi;
            bfrag[tt][kk][0] = Wf[(r + 0) * HC_ + jb];
            bfrag[tt][kk][1] = Wf[(r + 1) * HC_ + jb];
        }
    }

    // ---- init: X[:,0], X[:,1]; thread t owns (b,c) = (t>>4, t&15) ----
    const int tb = tid >> 4;
    const int tc = tid & 15;
    const float* Xrow = X + (size_t)(gb0 + tb) * L_ * C_ + tc;
    const float x0 = Xrow[0];
    const float x1 = Xrow[C_];
    xcur[tb * 17 + tc] = x0;
    __syncthreads();

    // z0 = X[:,0] @ W_init + b_init : wave w computes its 4 columns
    #pragma unroll
    for (int rep = 0; rep < 2; ++rep) {
        const int idx = lane + 32 * rep;
        const int b   = idx & 15;
        const int h   = 4 * w + (idx >> 4);
        float a0 = bi[h];
        #pragma unroll
        for (int c = 0; c < C_; ++c)
            a0 += xcur[b * 17 + c] * Wi[c * H_ + h];
        zb[b * ZS + h] = a0;
    }
    __syncthreads();

    {   // diff[0]; d[0]=diff[0] so ga(0)=gm(0)=gc(0)=diff[0]
        const float d0 = x1 - x0;
        xcur[tb * 17 + tc]     = x1;
        dbuf[0][tb * DXS + tc] = d0;
        dbuf[1][tb * DXS + tc] = d0;
        gmb[tb * DXS + tc]     = d0;    // 1.5*d0 - 0.25*(d0+d0)
    }
    float xnext = Xrow[2 * C_];         // software pipeline: X[:, t+2]
    __syncthreads();

    for (int t = 0; t < L_ - 1; ++t) {
        if (t >= 1) {   // step-start dX update (races excluded by barrier schedule)
            const float xn = xnext;
            const float dc = xn - xcur[tb * 17 + tc];
            const float dp = dbuf[(t + 1) & 1][tb * DXS + tc];
            xcur[tb * 17 + tc]           = xn;
            dbuf[t & 1][tb * DXS + tc]   = dc;
            gmb[tb * DXS + tc]           = 1.25f * dc - 0.25f * dp; // 1.5dc-.25(dp+dc)
            if (t < L_ - 2) xnext = Xrow[(size_t)(t + 2) * C_];
        }

        #pragma unroll
        for (int s = 0; s < 4; ++s) {
            // A fragments: z_eval = z + coef * k[s-1]
            const float coef = (s == 3) ? 1.0f : 0.5f;
            v2f a[8];
            #pragma unroll
            for (int kk = 0; kk < 8; ++kk) {
                const int off = lo * ZS + 4 * kk + 2 * hi;
                v2f zz = *(const v2f*)&zb[off];
                if (s > 0) {
                    v2f kv = *(const v2f*)&kb[s - 1][off];
                    zz[0] += coef * kv[0];
                    zz[1] += coef * kv[1];
                }
                a[kk] = zz;
            }

            // GEMM: 4 independent accumulator chains, C starts as inline 0
            v8f acc[4];
            #pragma unroll
            for (int tt = 0; tt < 4; ++tt)
                acc[tt] = (v8f){0.f, 0.f, 0.f, 0.f, 0.f, 0.f, 0.f, 0.f};
            #pragma unroll
            for (int kk = 0; kk < 8; ++kk)
                #pragma unroll
                for (int tt = 0; tt < 4; ++tt)
                    acc[tt] = __builtin_amdgcn_wmma_f32_16x16x4_f32(
                        false, a[kk], false, bfrag[tt][kk],
                        (short)0, acc[tt], false, false);

            // bias + tanh, stage F tile to per-wave LDS (same-wave, in-order DS)
            #pragma unroll
            for (int tt = 0; tt < 4; ++tt)
                #pragma unroll
                for (int v = 0; v < 8; ++v)
                    fbw[(v + 8 * hi) * FS + tt * 16 + lo] =
                        fast_tanh(acc[tt][v] + bfr[tt]);

            // stage dX: s0 -> ga(prev diff), s1/s2 -> gm, s3 -> gc(cur diff)
            const float* dxs = (s == 0) ? dbuf[(t + 1) & 1]
                             : (s == 3) ? dbuf[t & 1] : gmb;

            // contraction over c via b128 row reads; lane owns (b, h4) pairs,
            // writes k_s[b, h=4w+h4] directly in the z-update lane mapping
            #pragma unroll
            for (int rep = 0; rep < 2; ++rep) {
                const int idx = lane + 32 * rep;
                const int b   = idx & 15;
                const int h4  = idx >> 4;       // 0,1 (rep0) / 2,3 (rep1)
                const float4* frow = (const float4*)&fbw[b * FS + h4 * 16];
                const float4* drow = (const float4*)&dxs[b * DXS];
                float sum = 0.0f;
                #pragma unroll
                for (int q = 0; q < 4; ++q) {
                    const float4 fv = frow[q];
                    const float4 dv = drow[q];
                    sum += fv.x * dv.x + fv.y * dv.y + fv.z * dv.z + fv.w * dv.w;
                }
                kb[s][b * ZS + 4 * w + h4] = sum;
            }
            __syncthreads();
        }

        // z += (k1 + 2k2 + 2k3 + k4)/6 : wave w updates its 4 columns
        #pragma unroll
        for (int rep = 0; rep < 2; ++rep) {
            const int idx = lane + 32 * rep;
            const int b   = idx & 15;
            const int h   = 4 * w + (idx >> 4);
            const int o   = b * ZS + h;
            zb[o] += (kb[0][o] + 2.0f * kb[1][o] + 2.0f * kb[2][o] + kb[3][o])
                     * (1.0f / 6.0f);
        }
        __syncthreads();
    }

    // write zT (16 x 32 tile)
    #pragma unroll
    for (int rep = 0; rep < 2; ++rep) {
        const int idx = tid + 256 * rep;
        const int b   = idx >> 5;
        const int h   = idx & 31;
        out[(size_t)(gb0 + b) * H_ + h] = zb[b * ZS + h];
    }
}

extern "C" void kernel_launch(void* const* d_in, const int* in_sizes, int n_in,
                              void* d_out, int out_size, void* d_ws, size_t ws_size,
                              hipStream_t stream) {
    (void)in_sizes; (void)n_in; (void)out_size; (void)d_ws; (void)ws_size;
    const float* X  = (const float*)d_in[0];
    const float* Wi = (const float*)d_in[1];
    const float* bi = (const float*)d_in[2];
    const float* Wf = (const float*)d_in[3];
    const float* bf = (const float*)d_in[4];
    float* out = (float*)d_out;
    cde_rk4_kernel<<<B_ / 16, 256, 0, stream>>>(X, Wi, bi, Wf, bf, out);
}